// RepulsivePrior_85572928406158
// MI455X (gfx1250) — compile-verified
//
#include <hip/hip_runtime.h>
#include <hip/hip_bf16.h>
#include <math.h>

typedef __attribute__((ext_vector_type(2))) float v2f;
typedef __attribute__((ext_vector_type(8))) float v8f;

static constexpr int   BATCH   = 8;
static constexpr int   NATOM   = 8192;
static constexpr int   KNBR    = 100;
static constexpr int   QPB     = NATOM * KNBR / 4;  // 204800 quads per batch
static constexpr int   BLOCKS_PER_BATCH = 200;      // 200*256 threads -> 4 quads/thread
static constexpr int   THREADS = 256;               // 8 wave32s per block
static constexpr float R_MIN_F = 0.1f;
static constexpr float R_MAX_F = 2.0f;

__global__ void rp_zero(float* out, int n) {
    int i = blockIdx.x * blockDim.x + threadIdx.x;
    if (i < n) out[i] = 0.0f;
}

__global__ __launch_bounds__(THREADS) void rp_main(
    const float* __restrict__ positions,   // [B,N,3]   f32
    const float* __restrict__ cell,        // [B,3,3]   f32
    const float* __restrict__ offsets,     // [B,N,K,3] f32
    const long long* __restrict__ neighbors, // [B,N,K] i64
    const int* __restrict__ mask,          // [B,N,K]   i32
    float* __restrict__ out)               // [B]       f32
{
    const int b = blockIdx.y;

    // 3x3 cell, row-major [d][c]; uniform per block -> scalar loads
    const float* cb = cell + b * 9;
    const float c00 = cb[0], c01 = cb[1], c02 = cb[2];
    const float c10 = cb[3], c11 = cb[4], c12 = cb[5];
    const float c20 = cb[6], c21 = cb[7], c22 = cb[8];

    const float*  posb = positions + (size_t)b * NATOM * 3;
    const float4* off4 = (const float4*)(offsets + (size_t)b * NATOM * KNBR * 3);
    const int4*   nb4  = (const int4*)(neighbors + (size_t)b * NATOM * KNBR);
    const int4*   mb4  = (const int4*)(mask + (size_t)b * NATOM * KNBR);

    const int qstride = gridDim.x * blockDim.x;
    float acc = 0.0f;

    auto body = [&](int m, int j, float ox, float oy, float oz,
                    float pix, float piy, float piz) {
        const float* pj = posb + j * 3;
        float vx = pj[0] + ox * c00 + oy * c10 + oz * c20 - pix;
        float vy = pj[1] + ox * c01 + oy * c11 + oz * c21 - piy;
        float vz = pj[2] + ox * c02 + oy * c12 + oz * c22 - piz;
        float sq = vx * vx + vy * vy + vz * vz;
        float d  = sqrtf(sq);
        bool keep = (m > 0) && (d >= R_MIN_F) && (d <= R_MAX_F);
        acc += keep ? (1.0f / sq) : 0.0f;
    };

    for (int q = blockIdx.x * blockDim.x + threadIdx.x; q < QPB; q += qstride) {
        // fully coalesced 128-bit streaming loads (quad of 4 neighbor entries)
        float4 o0 = off4[3 * q + 0];
        float4 o1 = off4[3 * q + 1];
        float4 o2 = off4[3 * q + 2];
        int4   j0 = nb4[2 * q + 0];   // two int64 entries: lo dwords at .x/.z
        int4   j1 = nb4[2 * q + 1];
        int4   m4 = mb4[q];

        // speculative prefetch of next grid-stride iteration's stream
        __builtin_prefetch(&off4[3 * (q + qstride)], 0, 0);
        __builtin_prefetch(&nb4[2 * (q + qstride)], 0, 0);

        // K=100 is a multiple of 4 -> all 4 elements share atom row n
        int n = q / 25;                       // n = (4q)/100
        const float* pi = posb + n * 3;
        float pix = pi[0], piy = pi[1], piz = pi[2];

        body(m4.x, j0.x, o0.x, o0.y, o0.z, pix, piy, piz);
        body(m4.y, j0.z, o0.w, o1.x, o1.y, pix, piy, piz);
        body(m4.z, j1.x, o1.z, o1.w, o2.x, pix, piy, piz);
        body(m4.w, j1.z, o2.y, o2.z, o2.w, pix, piy, piz);
    }

    // --- wave32 reduction via V_WMMA_F32_16X16X4_F32 ---
    // A (16x4): each lane contributes its partial in VGPR0, zero in VGPR1.
    // B (4x16) = all ones, so D[m][n] = row-sum of A and
    // sum_m D[m][.] == sum of all 64 A values == wave sum of partials.
    v2f a;  a.x = acc;  a.y = 0.0f;
    v2f bo; bo.x = 1.0f; bo.y = 1.0f;
    v8f cz = {};
    cz = __builtin_amdgcn_wmma_f32_16x16x4_f32(
        /*neg_a=*/false, a, /*neg_b=*/false, bo,
        /*c_mod=*/(short)0, cz, /*reuse_a=*/false, /*reuse_b=*/false);

    // lanes 0..15 hold rows 0..7 (col=lane), lanes 16..31 hold rows 8..15
    float s = cz[0] + cz[1] + cz[2] + cz[3] + cz[4] + cz[5] + cz[6] + cz[7];
    s += __shfl_xor(s, 16, 32);               // combine the two half-row sums

    if ((threadIdx.x & 31) == 0)
        atomicAdd(&out[b], 0.5f * s);
}

extern "C" void kernel_launch(void* const* d_in, const int* in_sizes, int n_in,
                              void* d_out, int out_size, void* d_ws, size_t ws_size,
                              hipStream_t stream) {
    (void)in_sizes; (void)n_in; (void)d_ws; (void)ws_size;
    const float*     positions = (const float*)d_in[0];
    const float*     cell      = (const float*)d_in[1];
    const float*     offsets   = (const float*)d_in[2];
    const long long* neighbors = (const long long*)d_in[3];
    const int*       mask      = (const int*)d_in[4];
    float*           out       = (float*)d_out;

    rp_zero<<<1, 64, 0, stream>>>(out, out_size);

    dim3 grid(BLOCKS_PER_BATCH, BATCH);
    rp_main<<<grid, THREADS, 0, stream>>>(positions, cell, offsets,
                                          neighbors, mask, out);
}